// Quantizer_53480932770374
// MI455X (gfx1250) — compile-verified
//
#include <hip/hip_runtime.h>

typedef __attribute__((ext_vector_type(2))) float v2f;
typedef __attribute__((ext_vector_type(8))) float v8f;

#define N_ROWS 4096
#define DIMS   16
#define KCODES 65536
#define KSPLIT 16
#define CHUNKS_PER_SPLIT (KCODES / 16 / KSPLIT)   // 256 chunks of 16 codes

// ---------------------------------------------------------------------------
// Prep 1: transpose x [B=4, D=16, T=1024] -> xt [N=4096, D=16] row-major
// ---------------------------------------------------------------------------
__global__ __launch_bounds__(256) void vq_prep_transpose(const float* __restrict__ x,
                                                         float* __restrict__ xt) {
  int tid = blockIdx.x * blockDim.x + threadIdx.x;   // 0..65535
  int n = tid >> 4, d = tid & 15;
  int b = n >> 10, t = n & 1023;
  xt[tid] = x[(b * 16 + d) * 1024 + t];
}

// ---------------------------------------------------------------------------
// Prep 2: cnh[j] = 0.5 * |codebook[j]|^2
// ---------------------------------------------------------------------------
__global__ __launch_bounds__(256) void vq_prep_cnorm(const float* __restrict__ cb,
                                                     float* __restrict__ cnh) {
  int j = blockIdx.x * blockDim.x + threadIdx.x;     // 0..65535
  const float4* r = (const float4*)(cb + j * DIMS);
  float s = 0.f;
#pragma unroll
  for (int i = 0; i < 4; ++i) {
    float4 v = r[i];
    s += v.x * v.x + v.y * v.y + v.z * v.z + v.w * v.w;
  }
  cnh[j] = 0.5f * s;
}

// ---------------------------------------------------------------------------
// Main: one wave per (16-row tile, K-split). 4 chained V_WMMA_F32_16X16X4_F32
// per 16x16 tile of dot products; running argmax of (x.c - 0.5|c|^2) which is
// equivalent to argmin of squared distance. Software-pipelined: chunk c+1's
// loads are issued before chunk c's WMMA/argmax so they overlap.
// ---------------------------------------------------------------------------
__global__ __launch_bounds__(32) void vq_search(const float* __restrict__ xt,
                                                const float* __restrict__ cb,
                                                const float* __restrict__ cnh,
                                                float* __restrict__ pval,
                                                int* __restrict__ pidx) {
  const int lane = threadIdx.x;        // 0..31
  const int half = lane >> 4;          // 0: rows 0-7 of C/D, 1: rows 8-15
  const int l16  = lane & 15;          // column (code within chunk), row for A
  const int dimOff = half * 2;         // K-dim offset for A/B fragments
  const int rowTile = blockIdx.x;      // 0..255
  const int ks = blockIdx.y;           // 0..KSPLIT-1

  // A fragments: lane holds x[rowTile*16 + l16][kbase + dimOff .. +1]
  const float* xrow = xt + (rowTile * 16 + l16) * DIMS;
  v2f a0 = *(const v2f*)(xrow + 0  + dimOff);
  v2f a1 = *(const v2f*)(xrow + 4  + dimOff);
  v2f a2 = *(const v2f*)(xrow + 8  + dimOff);
  v2f a3 = *(const v2f*)(xrow + 12 + dimOff);

  float best[8];
  int   bidx[8];
#pragma unroll
  for (int r = 0; r < 8; ++r) { best[r] = -3.4e38f; bidx[r] = 0; }

  const int cBeg = ks * CHUNKS_PER_SPLIT;
  const int cEnd = cBeg + CHUNKS_PER_SPLIT;

  // Prologue: load chunk cBeg.
  {
  }
  int code = cBeg * 16 + l16;
  const float* crow = cb + code * DIMS;
  v2f b0 = *(const v2f*)(crow + 0  + dimOff);
  v2f b1 = *(const v2f*)(crow + 4  + dimOff);
  v2f b2 = *(const v2f*)(crow + 8  + dimOff);
  v2f b3 = *(const v2f*)(crow + 12 + dimOff);
  float ch = cnh[code];

  for (int c = cBeg; c < cEnd; ++c) {
    // Prefetch next chunk (clamped on the last iteration; redundant reload
    // keeps the loop branch-free and the loads in flight during compute).
    int cn = (c + 1 < cEnd) ? (c + 1) : c;
    int ncode = cn * 16 + l16;
    const float* nrow = cb + ncode * DIMS;
    v2f nb0 = *(const v2f*)(nrow + 0  + dimOff);
    v2f nb1 = *(const v2f*)(nrow + 4  + dimOff);
    v2f nb2 = *(const v2f*)(nrow + 8  + dimOff);
    v2f nb3 = *(const v2f*)(nrow + 12 + dimOff);
    float nch = cnh[ncode];

    v8f acc = {};
    acc = __builtin_amdgcn_wmma_f32_16x16x4_f32(false, a0, false, b0, (short)0, acc, false, false);
    acc = __builtin_amdgcn_wmma_f32_16x16x4_f32(false, a1, false, b1, (short)0, acc, false, false);
    acc = __builtin_amdgcn_wmma_f32_16x16x4_f32(false, a2, false, b2, (short)0, acc, false, false);
    acc = __builtin_amdgcn_wmma_f32_16x16x4_f32(false, a3, false, b3, (short)0, acc, false, false);

#pragma unroll
    for (int r = 0; r < 8; ++r) {
      float t = acc[r] - ch;                       // maximize x.c - 0.5|c|^2
      if (t > best[r]) { best[r] = t; bidx[r] = code; }
    }

    b0 = nb0; b1 = nb1; b2 = nb2; b3 = nb3; ch = nch; code = ncode;
  }

  // Argmax across the 16 lanes of each half (xor masks stay within the half).
#pragma unroll
  for (int m = 1; m < 16; m <<= 1) {
#pragma unroll
    for (int r = 0; r < 8; ++r) {
      float ov = __shfl_xor(best[r], m, 32);
      int   oi = __shfl_xor(bidx[r], m, 32);
      if (ov > best[r] || (ov == best[r] && oi < bidx[r])) { best[r] = ov; bidx[r] = oi; }
    }
  }

  if (l16 == 0) {
    int rbase = rowTile * 16 + half * 8;
#pragma unroll
    for (int r = 0; r < 8; ++r) {
      pval[(rbase + r) * KSPLIT + ks] = best[r];
      pidx[(rbase + r) * KSPLIT + ks] = bidx[r];
    }
  }
}

// ---------------------------------------------------------------------------
// Finalize: global argmax over K-splits per row, gather codebook row into
// [B, D, T] output, per-row commitment-loss contribution.
// ---------------------------------------------------------------------------
__global__ __launch_bounds__(256) void vq_finalize(const float* __restrict__ xt,
                                                   const float* __restrict__ cb,
                                                   const float* __restrict__ pval,
                                                   const int* __restrict__ pidx,
                                                   float* __restrict__ out,
                                                   float* __restrict__ rowloss) {
  int n = blockIdx.x * blockDim.x + threadIdx.x;   // 0..4095
  float best = -3.4e38f;
  int bi = 0;
#pragma unroll
  for (int k = 0; k < KSPLIT; ++k) {
    float v = pval[n * KSPLIT + k];
    int   i = pidx[n * KSPLIT + k];
    if (v > best || (v == best && i < bi)) { best = v; bi = i; }
  }
  const float* q  = cb + bi * DIMS;
  const float* xr = xt + n * DIMS;
  int b = n >> 10, t = n & 1023;
  float loss = 0.f;
#pragma unroll
  for (int d = 0; d < DIMS; ++d) {
    float qv = q[d];
    out[(b * 16 + d) * 1024 + t] = qv;             // straight-through fwd == quant
    float diff = qv - xr[d];
    loss += diff * diff;
  }
  rowloss[n] = loss;
}

// ---------------------------------------------------------------------------
// Deterministic loss reduction (single block).
// ---------------------------------------------------------------------------
__global__ __launch_bounds__(256) void vq_loss(const float* __restrict__ rowloss,
                                               float* __restrict__ out) {
  __shared__ float sm[256];
  int tid = threadIdx.x;
  float s = 0.f;
  for (int i = tid; i < N_ROWS; i += 256) s += rowloss[i];
  sm[tid] = s;
  __syncthreads();
  for (int off = 128; off > 0; off >>= 1) {
    if (tid < off) sm[tid] += sm[tid + off];
    __syncthreads();
  }
  if (tid == 0) out[N_ROWS * DIMS] = sm[0] / (float)(N_ROWS * DIMS);
}

extern "C" void kernel_launch(void* const* d_in, const int* in_sizes, int n_in,
                              void* d_out, int out_size, void* d_ws, size_t ws_size,
                              hipStream_t stream) {
  const float* x  = (const float*)d_in[0];   // [4, 16, 1024]
  const float* cb = (const float*)d_in[1];   // [65536, 16]
  float* out = (float*)d_out;                // 65536 quant + 1 loss

  float* ws      = (float*)d_ws;
  float* xt      = ws;                        // 65536 floats
  float* cnh     = ws + 65536;                // 65536 floats
  float* pval    = ws + 131072;               // 65536 floats
  int*   pidx    = (int*)(ws + 196608);       // 65536 ints
  float* rowloss = ws + 262144;               // 4096 floats

  vq_prep_transpose<<<256, 256, 0, stream>>>(x, xt);
  vq_prep_cnorm<<<256, 256, 0, stream>>>(cb, cnh);

  dim3 grid(256, KSPLIT);
  vq_search<<<grid, 32, 0, stream>>>(xt, cb, cnh, pval, pidx);

  vq_finalize<<<16, 256, 0, stream>>>(xt, cb, pval, pidx, out, rowloss);
  vq_loss<<<1, 256, 0, stream>>>(rowloss, out);
}